// TransformerDownSampling_77661598646727
// MI455X (gfx1250) — compile-verified
//
#include <hip/hip_runtime.h>
#include <hip/hip_bf16.h>

typedef float v2f __attribute__((ext_vector_type(2)));
typedef float v8f __attribute__((ext_vector_type(8)));

#define NPTS   16384
#define CIN    128
#define COUT   256
#define BATCH  8
#define NPOINT 4096

// ---------------------------------------------------------------------------
// Kernel 1: GEMM  y[b,o,n] = sum_c feat[b,n,c] * W[o,c] + bias[o]
// layout out: y is [B, COUT, NPTS] (channel-major for coalesced stats)
// One block = 4 waves, all share one n-tile (16 points).
// Each wave computes a 64(o) x 16(n) strip: 4 accumulators, K-loop of 32
// V_WMMA_F32_16X16X4_F32 ops each -> 128 wmma / wave.
// A (16x4 f32): lane l<16 holds row o0+l, K={0,1}; lane>=16 same row, K={2,3}.
// B (4x16 f32): lane l<16 holds col n0+l, K={0,1}; lane>=16 same col, K={2,3}.
// C/D (16x16 f32): vgpr v, lane l -> (o0 + v + 8*(l/16), n0 + l%16).
// ---------------------------------------------------------------------------
__global__ __launch_bounds__(128) void gemm_in_wmma(
    const float* __restrict__ feat,   // [B, NPTS, CIN]
    const float* __restrict__ Wm,     // [COUT, CIN]
    const float* __restrict__ bias,   // [COUT]
    float* __restrict__ y)            // [B, COUT, NPTS]
{
    const int wave  = threadIdx.x >> 5;
    const int lane  = threadIdx.x & 31;
    const int half  = lane >> 4;       // 0: K 0..1,  1: K 2..3
    const int l16   = lane & 15;

    const int blk   = blockIdx.x;      // B * (NPTS/16) = 8192 blocks
    const int b     = blk >> 10;
    const int n0    = (blk & 1023) << 4;
    const int obase = wave << 6;       // each wave: o in [obase, obase+64)

    const float* bptr = feat + ((size_t)(b * NPTS + n0 + l16)) * CIN + 2 * half;

    const float* aptr[4];
    v8f c[4];
#pragma unroll
    for (int j = 0; j < 4; ++j) {
        const int o0 = obase + j * 16;
        aptr[j] = Wm + (size_t)(o0 + l16) * CIN + 2 * half;
#pragma unroll
        for (int v = 0; v < 8; ++v)
            c[j][v] = bias[o0 + v + 8 * half];   // accumulate on top of bias
    }

#pragma unroll 4
    for (int k = 0; k < CIN / 4; ++k) {
        const v2f bb = *(const v2f*)(bptr + k * 4);
#pragma unroll
        for (int j = 0; j < 4; ++j) {
            const v2f a = *(const v2f*)(aptr[j] + k * 4);
            c[j] = __builtin_amdgcn_wmma_f32_16x16x4_f32(
                false, a, false, bb, (short)0, c[j], false, false);
        }
    }

#pragma unroll
    for (int j = 0; j < 4; ++j) {
        const int o0 = obase + j * 16;
#pragma unroll
        for (int v = 0; v < 8; ++v) {
            const int o = o0 + v + 8 * half;
            y[((size_t)(b * COUT + o)) * NPTS + n0 + l16] = c[j][v];
        }
    }
}

// ---------------------------------------------------------------------------
// Kernel 2: per-(b,o) mean / rstd over N (coalesced, y is channel-major)
// ---------------------------------------------------------------------------
__global__ __launch_bounds__(256) void instnorm_stats(
    const float* __restrict__ y, float* __restrict__ mean, float* __restrict__ rstd)
{
    const int bo = blockIdx.x;                 // 0 .. B*COUT-1
    const float* p = y + (size_t)bo * NPTS;

    float s = 0.f, s2 = 0.f;
    for (int n = threadIdx.x; n < NPTS; n += 256) {
        const float v = p[n];
        s += v; s2 += v * v;
    }
#pragma unroll
    for (int off = 16; off > 0; off >>= 1) {
        s  += __shfl_xor(s,  off, 32);
        s2 += __shfl_xor(s2, off, 32);
    }
    __shared__ float ls[8], ls2[8];
    const int wave = threadIdx.x >> 5, lane = threadIdx.x & 31;
    if (lane == 0) { ls[wave] = s; ls2[wave] = s2; }
    __syncthreads();
    if (threadIdx.x == 0) {
        float ts = 0.f, ts2 = 0.f;
        for (int w = 0; w < 8; ++w) { ts += ls[w]; ts2 += ls2[w]; }
        const float m   = ts  * (1.0f / NPTS);
        const float var = ts2 * (1.0f / NPTS) - m * m;
        mean[bo] = m;
        rstd[bo] = rsqrtf(var + 1e-5f);
    }
}

// ---------------------------------------------------------------------------
// Kernel 3: farthest point sampling. One workgroup per batch, 1024 threads,
// 16 points per thread held entirely in registers (coords + running dist).
// Per round: register distance update, wave shfl_xor argmax (carrying
// d/idx/x/y/z), LDS cross-wave reduce, 2 barriers. Tie-break: lowest index.
// ---------------------------------------------------------------------------
__global__ __launch_bounds__(1024) void fps_kernel(
    const float* __restrict__ points,   // [B, NPTS, 3]
    int*   __restrict__ fpsIdx,         // [B, NPOINT]
    float* __restrict__ outPts)         // [B, NPOINT, 3]
{
    const int T = 1024, PPT = NPTS / 1024;
    const int b = blockIdx.x;
    const int tid = threadIdx.x;
    const int wave = tid >> 5, lane = tid & 31;

    __shared__ float red[32 * 5];     // per-wave best: d, idx(bits), x, y, z
    __shared__ float bcast[4];        // winner x, y, z (+ pad)

    float px[PPT], py[PPT], pz[PPT], dist[PPT];
    const float* base = points + (size_t)b * NPTS * 3;

    float sx = 0.f, sy = 0.f, sz = 0.f;
#pragma unroll
    for (int i = 0; i < PPT; ++i) {
        const int n = tid + i * T;
        px[i] = base[n * 3 + 0];
        py[i] = base[n * 3 + 1];
        pz[i] = base[n * 3 + 2];
        sx += px[i]; sy += py[i]; sz += pz[i];
        dist[i] = 1e10f;
    }
    // --- centroid reduction (reuse `red` as 32x3 scratch) ---
#pragma unroll
    for (int off = 16; off > 0; off >>= 1) {
        sx += __shfl_xor(sx, off, 32);
        sy += __shfl_xor(sy, off, 32);
        sz += __shfl_xor(sz, off, 32);
    }
    if (lane == 0) { red[wave * 3 + 0] = sx; red[wave * 3 + 1] = sy; red[wave * 3 + 2] = sz; }
    __syncthreads();
    if (tid == 0) {
        float tx = 0.f, ty = 0.f, tz = 0.f;
        for (int w = 0; w < 32; ++w) {
            tx += red[w * 3 + 0]; ty += red[w * 3 + 1]; tz += red[w * 3 + 2];
        }
        bcast[0] = tx * (1.0f / NPTS);
        bcast[1] = ty * (1.0f / NPTS);
        bcast[2] = tz * (1.0f / NPTS);
    }
    __syncthreads();
    float cx = bcast[0], cy = bcast[1], cz = bcast[2];

    // --- 4096 sequential rounds ---
    for (int s = 0; s < NPOINT; ++s) {
        float bd = -1.0f, bx = 0.f, by = 0.f, bz = 0.f;
        int   bi = 0x7fffffff;
#pragma unroll
        for (int i = 0; i < PPT; ++i) {
            const float dx = px[i] - cx, dy = py[i] - cy, dz = pz[i] - cz;
            float d = dx * dx + dy * dy + dz * dz;
            if (s != 0) {                       // round 0: argmax vs centroid only
                d = fminf(dist[i], d);
                dist[i] = d;
            }
            const int n = tid + i * T;
            if (d > bd || (d == bd && n < bi)) {
                bd = d; bi = n; bx = px[i]; by = py[i]; bz = pz[i];
            }
        }
        // wave argmax
#pragma unroll
        for (int off = 16; off > 0; off >>= 1) {
            const float od = __shfl_xor(bd, off, 32);
            const int   oi = __shfl_xor(bi, off, 32);
            const float ox = __shfl_xor(bx, off, 32);
            const float oy = __shfl_xor(by, off, 32);
            const float oz = __shfl_xor(bz, off, 32);
            if (od > bd || (od == bd && oi < bi)) {
                bd = od; bi = oi; bx = ox; by = oy; bz = oz;
            }
        }
        if (lane == 0) {
            red[wave * 5 + 0] = bd;
            ((int*)red)[wave * 5 + 1] = bi;
            red[wave * 5 + 2] = bx; red[wave * 5 + 3] = by; red[wave * 5 + 4] = bz;
        }
        __syncthreads();
        if (wave == 0) {
            bd = red[lane * 5 + 0];
            bi = ((const int*)red)[lane * 5 + 1];
            bx = red[lane * 5 + 2]; by = red[lane * 5 + 3]; bz = red[lane * 5 + 4];
#pragma unroll
            for (int off = 16; off > 0; off >>= 1) {
                const float od = __shfl_xor(bd, off, 32);
                const int   oi = __shfl_xor(bi, off, 32);
                const float ox = __shfl_xor(bx, off, 32);
                const float oy = __shfl_xor(by, off, 32);
                const float oz = __shfl_xor(bz, off, 32);
                if (od > bd || (od == bd && oi < bi)) {
                    bd = od; bi = oi; bx = ox; by = oy; bz = oz;
                }
            }
            if (lane == 0) {
                bcast[0] = bx; bcast[1] = by; bcast[2] = bz;
                fpsIdx[b * NPOINT + s] = bi;
                float* op = outPts + ((size_t)b * NPOINT + s) * 3;
                op[0] = bx; op[1] = by; op[2] = bz;
            }
        }
        __syncthreads();
        cx = bcast[0]; cy = bcast[1]; cz = bcast[2];
    }
}

// ---------------------------------------------------------------------------
// Kernel 4: gather sampled features with lazy InstanceNorm + ReLU.
// Block = one sampled point, 256 threads = 256 channels.
// ---------------------------------------------------------------------------
__global__ __launch_bounds__(256) void gather_feats(
    const float* __restrict__ y,      // [B, COUT, NPTS]
    const float* __restrict__ mean,
    const float* __restrict__ rstd,
    const int*   __restrict__ fpsIdx, // [B, NPOINT]
    float* __restrict__ outF)         // [B, NPOINT, COUT]
{
    const int blk = blockIdx.x;          // b * NPOINT + s
    const int b = blk >> 12;             // NPOINT = 4096
    const int s = blk & (NPOINT - 1);
    const int o = threadIdx.x;

    const int n = fpsIdx[b * NPOINT + s];
    const int bo = b * COUT + o;
    float v = y[(size_t)bo * NPTS + n];
    v = (v - mean[bo]) * rstd[bo];
    v = fmaxf(v, 0.0f);
    outF[((size_t)b * NPOINT + s) * COUT + o] = v;
}

// ---------------------------------------------------------------------------
extern "C" void kernel_launch(void* const* d_in, const int* in_sizes, int n_in,
                              void* d_out, int out_size, void* d_ws, size_t ws_size,
                              hipStream_t stream) {
    (void)in_sizes; (void)n_in; (void)out_size; (void)ws_size;

    const float* points = (const float*)d_in[0];   // [8,16384,3]
    const float* feats  = (const float*)d_in[1];   // [8,16384,128]
    const float* Wm     = (const float*)d_in[2];   // [256,128]
    const float* bias   = (const float*)d_in[3];   // [256]
    // d_in[4] = num_points (device scalar); grid shapes must be host-known -> 4096

    float* out      = (float*)d_out;
    float* outPts   = out;                                   // [8,4096,3]
    float* outFeats = out + (size_t)BATCH * NPOINT * 3;      // [8,4096,256]

    char* ws = (char*)d_ws;
    float* y    = (float*)ws;                                        // 128 MB
    float* mean = (float*)(ws + (size_t)BATCH * COUT * NPTS * 4);
    float* rstd = mean + BATCH * COUT;
    int*   idx  = (int*)(rstd + BATCH * COUT);

    // 1) GEMM + bias via f32 WMMA
    gemm_in_wmma<<<BATCH * (NPTS / 16), 128, 0, stream>>>(feats, Wm, bias, y);
    // 2) InstanceNorm statistics
    instnorm_stats<<<BATCH * COUT, 256, 0, stream>>>(y, mean, rstd);
    // 3) Farthest point sampling (writes sampled points directly)
    fps_kernel<<<BATCH, 1024, 0, stream>>>(points, idx, outPts);
    // 4) Gather + normalize + ReLU
    gather_feats<<<BATCH * NPOINT, 256, 0, stream>>>(y, mean, rstd, idx, outFeats);
}